// Damping_19868518711856
// MI455X (gfx1250) — compile-verified
//
#include <hip/hip_runtime.h>
#include <stdint.h>

// ---------------------------------------------------------------------------
// Damping recurrence: y[0]=f[0]; y[i] = d*(y[i-1] + f[i]), d per-channel const.
// Memory-bound (512MB @ 23.3TB/s ~ 22us). Strategy:
//   - 1 block (256 thr, 8 wave32) per row of T=4096 fp32, grid=(C,B) so the
//     channel index needs no integer division
//   - async global->LDS B128 copy (ASYNCcnt, NT hint) for the row,
//     bank-conflict-free padded LDS layout (base banks tile 64 banks stride-4)
//   - per-thread 16-element serial scan in registers,
//     wave32 shuffle scan of segment carries (ratio d^16, squared per step),
//     LDS for the 8 cross-wave carries,
//   - non-temporal B128 stores of the result (write-once stream).
// ---------------------------------------------------------------------------

#define T_LEN   4096
#define BLOCK   256
#define SEG     16          // elements per thread
#define SEG_PAD 20          // floats per segment in LDS (16 data + 4 pad)
#define WAVES   (BLOCK / 32)
#define CHUNKS_PER_THREAD (T_LEN / 4 / BLOCK)   // 16B chunks each thread copies

typedef float f4  __attribute__((ext_vector_type(4)));
typedef int   i4  __attribute__((ext_vector_type(4)));
typedef i4 __attribute__((address_space(1)))* gload_ptr;   // global src (non-const)
typedef i4 __attribute__((address_space(3)))* lds_ptr;     // LDS dst

#if defined(__has_builtin)
#if __has_builtin(__builtin_amdgcn_global_load_async_to_lds_b128)
#define HAVE_ASYNC_B128 1
#endif
#if __has_builtin(__builtin_amdgcn_s_wait_asynccnt)
#define HAVE_WAIT_ASYNC 1
#endif
#endif
#ifndef HAVE_ASYNC_B128
#define HAVE_ASYNC_B128 0
#endif
#ifndef HAVE_WAIT_ASYNC
#define HAVE_WAIT_ASYNC 0
#endif

__global__ __launch_bounds__(BLOCK) void damping_scan_kernel(
    const float* __restrict__ forces,
    const float* __restrict__ damping_param,
    float* __restrict__ out) {
  __shared__ float smem[BLOCK * SEG_PAD];   // 20 KB row staging (padded)
  __shared__ float waveTot[WAVES];

  const int c    = blockIdx.x;                        // channel
  const int row  = blockIdx.y * gridDim.x + c;        // b*C + c
  const int t    = threadIdx.x;
  const int lane = t & 31;
  const int wave = t >> 5;

  const float* rowIn  = forces + (size_t)row * T_LEN;
  float*       rowOut = out    + (size_t)row * T_LEN;

  // d = 0.5 + sigmoid(p) * (0.9999 - 0.5), uniform per block
  const float p = damping_param[c];
  const float sig = 1.0f / (1.0f + expf(-p));
  const float d = fmaf(sig, 0.4999f, 0.5f);

  // ---- stage the row into LDS: async global->LDS, coalesced 16B chunks ----
  // chunk i holds elements [4i,4i+4); segment s=i>>2 quarter q=i&3 lands at
  // padded LDS offset s*SEG_PAD + 4q (keeps ds_load_b128 bank-conflict-free).
#pragma unroll
  for (int it = 0; it < CHUNKS_PER_THREAD; ++it) {
    const int chunk = it * BLOCK + t;
    const int s = chunk >> 2;
    const int q = chunk & 3;
    float* ldsp = &smem[s * SEG_PAD + q * 4];
    const float* gp = rowIn + (chunk << 2);
#if HAVE_ASYNC_B128
    // cpol=1 -> TH_LOAD_NT: input is a read-once stream, keep it out of L2
    __builtin_amdgcn_global_load_async_to_lds_b128(
        (gload_ptr)(void*)gp,
        (lds_ptr)(void*)ldsp,
        0, 1);
#else
    unsigned la = (unsigned)(uintptr_t)(void*)ldsp;
    unsigned long long ga = (unsigned long long)(uintptr_t)(void*)gp;
    asm volatile("global_load_async_to_lds_b128 %0, %1, off th:TH_LOAD_NT"
                 :: "v"(la), "v"(ga) : "memory");
#endif
  }
#if HAVE_WAIT_ASYNC
  __builtin_amdgcn_s_wait_asynccnt(0);
#else
  asm volatile("s_wait_asynccnt 0" ::: "memory");
#endif
  __syncthreads();

  // ---- per-thread local scan of 16 contiguous elements (zero carry-in) ----
  const f4* segp = (const f4*)&smem[t * SEG_PAD];
  const f4 v0 = segp[0];
  const f4 v1 = segp[1];
  const f4 v2 = segp[2];
  const f4 v3 = segp[3];

  float x[SEG];
#pragma unroll
  for (int i = 0; i < 4; ++i) {
    x[i]      = v0[i];
    x[4 + i]  = v1[i];
    x[8 + i]  = v2[i];
    x[12 + i] = v3[i];
  }

  float acc = 0.0f;
#pragma unroll
  for (int j = 0; j < SEG; ++j) {
    float g = d * x[j];
    if (j == 0 && t == 0) g = x[0];       // y[0] = f[0] at the row start
    acc = fmaf(d, acc, g);                // local[j] = g + d*local[j-1]
    x[j] = acc;                           // keep local scan in registers
  }
  const float lastLocal = acc;

  // dL = d^16 (ratio between consecutive thread segments)
  float dL = d * d;  dL *= dL;  dL *= dL;  dL *= dL;

  // ---- wave32 inclusive scan of segment carries with ratio dL ----
  float inc = lastLocal;
  float pw = dL;
#pragma unroll
  for (int k = 1; k < 32; k <<= 1) {
    float up = __shfl_up(inc, k, 32);
    if (lane >= k) inc = fmaf(pw, up, inc);
    pw *= pw;                              // pw ends as dL^32
  }
  const float dL32 = pw;

  if (lane == 31) waveTot[wave] = inc;
  __syncthreads();

  // carry entering this wave: wc_{w+1} = dL^32 * wc_w + waveTot_w
  float wc = 0.0f;
  for (int u = 0; u < wave; ++u) wc = fmaf(dL32, wc, waveTot[u]);

  // exclusive within-wave carry
  float excl = __shfl_up(inc, 1, 32);
  if (lane == 0) excl = 0.0f;

  // dL^lane by binary powering
  float plane = 1.0f, bb = dL;
#pragma unroll
  for (int k = 1; k < 32; k <<= 1) {
    if (lane & k) plane *= bb;
    bb *= bb;
  }
  const float carry = fmaf(plane, wc, excl);   // y just before this segment

  // ---- apply carry: y[j] = local[j] + d^{j+1} * carry, stream out NT ----
  float dp = d;
#pragma unroll
  for (int j = 0; j < SEG; ++j) {
    x[j] = fmaf(dp, carry, x[j]);
    dp *= d;
  }

  f4 o0, o1, o2, o3;
#pragma unroll
  for (int i = 0; i < 4; ++i) {
    o0[i] = x[i];
    o1[i] = x[4 + i];
    o2[i] = x[8 + i];
    o3[i] = x[12 + i];
  }
  f4* op = (f4*)(rowOut + t * SEG);
  __builtin_nontemporal_store(o0, op + 0);
  __builtin_nontemporal_store(o1, op + 1);
  __builtin_nontemporal_store(o2, op + 2);
  __builtin_nontemporal_store(o3, op + 3);
}

extern "C" void kernel_launch(void* const* d_in, const int* in_sizes, int n_in,
                              void* d_out, int out_size, void* d_ws, size_t ws_size,
                              hipStream_t stream) {
  const float* forces = (const float*)d_in[0];          // (B, C, T) fp32
  const float* damp   = (const float*)d_in[1];          // (1, C, 1) fp32
  float* out = (float*)d_out;
  const int C = in_sizes[1];                            // 1024
  const int rows = in_sizes[0] / T_LEN;                 // B*C = 16384
  const int B = rows / C;                               // 16
  dim3 grid(C, B, 1);
  damping_scan_kernel<<<grid, BLOCK, 0, stream>>>(forces, damp, out);
}